// HomogeneousGNN_55430847922243
// MI455X (gfx1250) — compile-verified
//
#include <hip/hip_runtime.h>
#include <hip/hip_bf16.h>
#include <stdint.h>

#define H 128            // hidden dim
#define NCLS 8
#define NGRAPH 64

typedef __attribute__((ext_vector_type(2))) float v2f;
typedef __attribute__((ext_vector_type(8))) float v8f;
typedef __attribute__((ext_vector_type(4))) unsigned int v4u;
typedef __attribute__((ext_vector_type(4))) int v4i;
typedef __attribute__((ext_vector_type(8))) int v8i;

// Tensor Data Mover availability (arity differs across toolchains)
#if defined(__HIP_DEVICE_COMPILE__) && \
    __has_builtin(__builtin_amdgcn_tensor_load_to_lds) && \
    __has_builtin(__builtin_amdgcn_s_wait_tensorcnt)
#define USE_TDM 1
#else
#define USE_TDM 0
#endif

// ---------------------------------------------------------------------------
// Stage the 128x128 f32 weight matrix into LDS.
// Preferred path: one TDM descriptor (2D tile copy) issued by wave 0; the DMA
// engine moves 64KB while the waves sit at the barrier, keeping the VMEM path
// free. Fallback: cooperative b128 loads.
// ---------------------------------------------------------------------------
__device__ __forceinline__ void stage_W(const float* __restrict__ W, float* Wl) {
#if USE_TDM
    if (threadIdx.x < 32) {           // TDM is a wave-level op; issue once
        uint64_t ga  = (uint64_t)(uintptr_t)W;
        uint32_t lds = (uint32_t)(uintptr_t)Wl;   // low 32 bits = LDS byte offset
        v4u g0;
        g0[0] = 1u;                                   // count=1, user-mode D#
        g0[1] = lds;                                  // lds_addr (bytes)
        g0[2] = (uint32_t)ga;                         // global_addr[31:0]
        g0[3] = (uint32_t)(ga >> 32) | (2u << 30);    // global_addr[56:32] | type=2
        v8i g1;
        g1[0] = (int)(2u << 16);                      // data_size = 4 bytes
        g1[1] = (int)(128u << 16);                    // tensor_dim0 = 128 (bits 63:48)
        g1[2] = (int)(128u << 16);                    // tensor_dim1 = 128 (bits 111:80 lo)
        g1[3] = (int)(128u << 16);                    // tile_dim0  = 128 (bits 127:112)
        g1[4] = 128;                                  // tile_dim1  = 128 (bits 143:128)
        g1[5] = 128;                                  // tensor_dim0_stride = 128
        g1[6] = (int)(16384u << 16);                  // tensor_dim1_stride lo16
        g1[7] = 0;
        v4i gz = {0, 0, 0, 0};
#if __clang_major__ >= 23
        v8i gz8 = {0, 0, 0, 0, 0, 0, 0, 0};
        __builtin_amdgcn_tensor_load_to_lds(g0, g1, gz, gz, gz8, 0);
#else
        __builtin_amdgcn_tensor_load_to_lds(g0, g1, gz, gz, 0);
#endif
        __builtin_amdgcn_s_wait_tensorcnt(0);         // s_wait_tensorcnt 0
    }
    __syncthreads();
#else
    const float4* Wg = (const float4*)W;
    float4* Ws = (float4*)Wl;
    #pragma unroll
    for (int i = 0; i < (H * H / 4) / 256; ++i)
        Ws[threadIdx.x + i * 256] = Wg[threadIdx.x + i * 256];
    __syncthreads();
#endif
}

// ---------------------------------------------------------------------------
// Degree / normalization setup
// ---------------------------------------------------------------------------
__global__ __launch_bounds__(256) void k_init_deg(float* deg, int n) {
    int t = blockIdx.x * 256 + threadIdx.x;
    if (t < n) deg[t] = 1.0f;                       // self-loop contributes 1
}

__global__ __launch_bounds__(256) void k_deg_accum(const int* __restrict__ dst,
                                                   float* deg, int nEdges) {
    int e = blockIdx.x * 256 + threadIdx.x;
    if (e < nEdges) atomicAdd(&deg[dst[e]], 1.0f);
}

__global__ __launch_bounds__(256) void k_rsqrt(float* deg, int n) {
    int t = blockIdx.x * 256 + threadIdx.x;
    if (t < n) deg[t] = rsqrtf(deg[t]);             // in place: deg -> dinv
}

__global__ __launch_bounds__(256) void k_edge_norm(const int* __restrict__ src,
                                                   const int* __restrict__ dst,
                                                   const float* __restrict__ dinv,
                                                   float* __restrict__ norm, int nEdges) {
    int e = blockIdx.x * 256 + threadIdx.x;
    if (e < nEdges) norm[e] = dinv[src[e]] * dinv[dst[e]];
}

// ---------------------------------------------------------------------------
// xw = (relu?)(X) @ W  via V_WMMA_F32_16X16X4_F32.
// 256 threads = 8 waves; each wave: one 16-row strip x all 128 cols.
// Per K-step: preload all 8 B-fragments (one DS clause), then 8 WMMAs
// back-to-back while the next A global_load_b64 is in flight.
// ---------------------------------------------------------------------------
__global__ __launch_bounds__(256) void k_gemm_wmma(const float* __restrict__ X,
                                                   const float* __restrict__ W,
                                                   float* __restrict__ Y,
                                                   int nRows, int applyRelu) {
    __shared__ float Wl[H * H];
    stage_W(W, Wl);

    const int wave  = threadIdx.x >> 5;
    const int lane  = threadIdx.x & 31;
    const int laneM = lane & 15;        // row within tile (A) / col (B,C,D)
    const int laneH = lane >> 4;        // lane-half selects K pair / M half
    const int rbase = (blockIdx.x * 8 + wave) * 16;
    if (rbase >= nRows) return;

    v8f acc[8] = {};

    int arow = rbase + laneM;
    if (arow >= nRows) arow = nRows - 1;            // safe clamp (N%16==0 anyway)
    const float* xrow = X + (size_t)arow * H + laneH * 2;

    #pragma unroll 2
    for (int kt = 0; kt < H / 4; ++kt) {
        // A fragment: lane holds A[laneM][2*laneH + v], v=0,1 -> contiguous b64
        v2f a = *(const v2f*)(xrow + kt * 4);
        if (applyRelu) { a.x = fmaxf(a.x, 0.0f); a.y = fmaxf(a.y, 0.0f); }
        const int kbase = kt * 4 + laneH * 2;
        v2f bf[8];
        #pragma unroll
        for (int nt = 0; nt < 8; ++nt) {
            bf[nt].x = Wl[(kbase + 0) * H + nt * 16 + laneM];
            bf[nt].y = Wl[(kbase + 1) * H + nt * 16 + laneM];
        }
        #pragma unroll
        for (int nt = 0; nt < 8; ++nt)
            acc[nt] = __builtin_amdgcn_wmma_f32_16x16x4_f32(
                false, a, false, bf[nt], (short)0, acc[nt], false, false);
    }

    // D layout: VGPR r, lanes 0-15 -> M=r, lanes 16-31 -> M=r+8; N=laneM
    #pragma unroll
    for (int nt = 0; nt < 8; ++nt) {
        #pragma unroll
        for (int r = 0; r < 8; ++r) {
            int row = rbase + r + 8 * laneH;
            if (row < nRows)
                Y[(size_t)row * H + nt * 16 + laneM] = acc[nt][r];
        }
    }
}

// ---------------------------------------------------------------------------
// y[n][d] = dinv[n]^2 * xw[n][d] + bias[d]   (self-loop + bias folded in)
// ---------------------------------------------------------------------------
__global__ __launch_bounds__(256) void k_init_y(const float* __restrict__ xw,
                                                const float* __restrict__ dinv,
                                                const float* __restrict__ bias,
                                                float* __restrict__ y, int nNodes) {
    int t = blockIdx.x * 256 + threadIdx.x;
    int n = t >> 7, d = t & (H - 1);
    if (n < nNodes) {
        float di = dinv[n];
        y[(size_t)n * H + d] = di * di * xw[(size_t)n * H + d] + bias[d];
    }
}

// ---------------------------------------------------------------------------
// Edge scatter: one wave32 per edge; each lane moves one float4 (32*4 = 128).
// Destination rows live in the 192MB L2, so the f32 atomics stay on-chip.
// ---------------------------------------------------------------------------
__global__ __launch_bounds__(256) void k_scatter(const int* __restrict__ src,
                                                 const int* __restrict__ dst,
                                                 const float* __restrict__ norm,
                                                 const float* __restrict__ XW,
                                                 float* __restrict__ Y, int nEdges) {
    int wave = threadIdx.x >> 5;
    int lane = threadIdx.x & 31;
    int e = blockIdx.x * 8 + wave;
    if (e >= nEdges) return;
    int s = src[e];
    int d = dst[e];
    float w = norm[e];
    if (lane == 0 && e + 8 < nEdges)        // global_prefetch_b8 next source row
        __builtin_prefetch(XW + (size_t)src[e + 8] * H, 0, 1);
    float4 v = ((const float4*)(XW + (size_t)s * H))[lane];
    float* yd = Y + (size_t)d * H + lane * 4;
    atomicAdd(yd + 0, w * v.x);
    atomicAdd(yd + 1, w * v.y);
    atomicAdd(yd + 2, w * v.z);
    atomicAdd(yd + 3, w * v.w);
}

// ---------------------------------------------------------------------------
// node_logits = emb @ clf_w + clf_b    (128x8, trivial)
// ---------------------------------------------------------------------------
__global__ __launch_bounds__(256) void k_logits(const float* __restrict__ emb,
                                                const float* __restrict__ Wc,
                                                const float* __restrict__ bc,
                                                float* __restrict__ out, int nNodes) {
    int t = blockIdx.x * 256 + threadIdx.x;
    int n = t >> 3, c = t & (NCLS - 1);
    if (n >= nNodes) return;
    const float* e = emb + (size_t)n * H;
    float a = bc[c];
    #pragma unroll 8
    for (int k = 0; k < H; ++k) a = fmaf(e[k], Wc[k * NCLS + c], a);
    out[(size_t)n * NCLS + c] = a;
}

// ---------------------------------------------------------------------------
// Graph pooling + regression head
// ---------------------------------------------------------------------------
__global__ __launch_bounds__(256) void k_pool_zero(float* gsum, float* gcnt) {
    int t = blockIdx.x * 256 + threadIdx.x;
    if (t < NGRAPH * H) gsum[t] = 0.0f;
    if (t < NGRAPH) gcnt[t] = 0.0f;
}

__global__ __launch_bounds__(256) void k_pool_accum(const float* __restrict__ emb,
                                                    const int* __restrict__ batch,
                                                    float* gsum, float* gcnt, int nNodes) {
    int t = blockIdx.x * 256 + threadIdx.x;
    int n = t >> 7, d = t & (H - 1);
    if (n >= nNodes) return;
    int g = batch[n];
    atomicAdd(&gsum[g * H + d], emb[(size_t)n * H + d]);
    if (d == 0) atomicAdd(&gcnt[g], 1.0f);
}

__global__ __launch_bounds__(128) void k_graph_head(const float* __restrict__ gsum,
                                                    const float* __restrict__ gcnt,
                                                    const float* __restrict__ w1,
                                                    const float* __restrict__ b1,
                                                    const float* __restrict__ w2,
                                                    const float* __restrict__ b2,
                                                    float* __restrict__ pred) {
    __shared__ float ge[H];
    __shared__ float hs[H / 2];
    int g = blockIdx.x;
    float inv = 1.0f / fmaxf(gcnt[g], 1.0f);
    ge[threadIdx.x] = gsum[g * H + threadIdx.x] * inv;
    __syncthreads();
    if (threadIdx.x < H / 2) {
        float a = b1[threadIdx.x];
        for (int k = 0; k < H; ++k) a = fmaf(ge[k], w1[k * (H / 2) + threadIdx.x], a);
        hs[threadIdx.x] = fmaxf(a, 0.0f);
    }
    __syncthreads();
    if (threadIdx.x == 0) {
        float a = b2[0];
        for (int j = 0; j < H / 2; ++j) a = fmaf(hs[j], w2[j], a);
        pred[g] = a;
    }
}

// ---------------------------------------------------------------------------
extern "C" void kernel_launch(void* const* d_in, const int* in_sizes, int n_in,
                              void* d_out, int out_size, void* d_ws, size_t ws_size,
                              hipStream_t stream) {
    const float* x      = (const float*)d_in[0];
    const int*   ei     = (const int*)d_in[1];
    const int*   batch  = (const int*)d_in[2];
    const float* conv_w = (const float*)d_in[3];
    const float* conv_b = (const float*)d_in[4];
    const float* clf_w  = (const float*)d_in[5];
    const float* clf_b  = (const float*)d_in[6];
    const float* reg_w1 = (const float*)d_in[7];
    const float* reg_b1 = (const float*)d_in[8];
    const float* reg_w2 = (const float*)d_in[9];
    const float* reg_b2 = (const float*)d_in[10];

    const int nNodes = in_sizes[2];          // batch length == N
    const int nEdges = in_sizes[1] / 2;      // edge_index is (2, E)
    const int* src = ei;
    const int* dst = ei + nEdges;

    // output layout: node_logits [N*8] | graph_pred [64] | node_embeddings [N*128]
    float* out_logits = (float*)d_out;
    float* out_pred   = out_logits + (size_t)nNodes * NCLS;
    float* out_emb    = out_pred + NGRAPH;

    // workspace layout (floats)
    float* wsf  = (float*)d_ws;
    size_t off  = 0;
    float* dinv = wsf + off; off += (size_t)nNodes;            // deg -> dinv
    float* norm = wsf + off; off += (size_t)nEdges;
    float* xw   = wsf + off; off += (size_t)nNodes * H;
    float* bufA = wsf + off; off += (size_t)nNodes * H;
    float* bufB = wsf + off; off += (size_t)nNodes * H;
    float* gsum = wsf + off; off += (size_t)NGRAPH * H;
    float* gcnt = wsf + off; off += (size_t)NGRAPH;

    const int TB = 256;
    const int gN    = (nNodes + TB - 1) / TB;
    const int gE    = (nEdges + TB - 1) / TB;
    const int gNH   = (int)(((size_t)nNodes * H + TB - 1) / TB);
    const int gGemm = (nNodes + 127) / 128;       // 8 waves x 16 rows per block
    const int gScat = (nEdges + 7) / 8;           // 1 wave per edge

    // --- normalization setup ---
    k_init_deg<<<gN, TB, 0, stream>>>(dinv, nNodes);
    k_deg_accum<<<gE, TB, 0, stream>>>(dst, dinv, nEdges);
    k_rsqrt<<<gN, TB, 0, stream>>>(dinv, nNodes);
    k_edge_norm<<<gE, TB, 0, stream>>>(src, dst, dinv, norm, nEdges);

    // --- layer 0 ---
    k_gemm_wmma<<<gGemm, TB, 0, stream>>>(x, conv_w, xw, nNodes, 0);
    k_init_y<<<gNH, TB, 0, stream>>>(xw, dinv, conv_b, bufA, nNodes);
    k_scatter<<<gScat, TB, 0, stream>>>(src, dst, norm, xw, bufA, nEdges);
    // --- layer 1 (relu of layer-0 output folded into A-load) ---
    k_gemm_wmma<<<gGemm, TB, 0, stream>>>(bufA, conv_w + H * H, xw, nNodes, 1);
    k_init_y<<<gNH, TB, 0, stream>>>(xw, dinv, conv_b + H, bufB, nNodes);
    k_scatter<<<gScat, TB, 0, stream>>>(src, dst, norm, xw, bufB, nEdges);
    // --- layer 2 (no relu on output; writes embeddings straight into d_out) ---
    k_gemm_wmma<<<gGemm, TB, 0, stream>>>(bufB, conv_w + 2 * H * H, xw, nNodes, 1);
    k_init_y<<<gNH, TB, 0, stream>>>(xw, dinv, conv_b + 2 * H, out_emb, nNodes);
    k_scatter<<<gScat, TB, 0, stream>>>(src, dst, norm, xw, out_emb, nEdges);

    // --- heads ---
    k_logits<<<(int)(((size_t)nNodes * NCLS + TB - 1) / TB), TB, 0, stream>>>(
        out_emb, clf_w, clf_b, out_logits, nNodes);
    k_pool_zero<<<(NGRAPH * H + TB - 1) / TB, TB, 0, stream>>>(gsum, gcnt);
    k_pool_accum<<<gNH, TB, 0, stream>>>(out_emb, batch, gsum, gcnt, nNodes);
    k_graph_head<<<NGRAPH, 128, 0, stream>>>(gsum, gcnt, reg_w1, reg_b1,
                                             reg_w2, reg_b2, out_pred);
}